// DoRA_APN_20057497273085
// MI455X (gfx1250) — compile-verified
//
#include <hip/hip_runtime.h>
#include <hip/hip_bf16.h>
#include <math.h>

// ---- problem constants (from reference) ----
#define Bb   4
#define NSs  8
#define Cc   3
#define HTt  224
#define HSs  384
#define Pp   16
#define EMB  768
#define NHh  12
#define HDd  64
#define Kk   12
#define TZz  128
#define NTt  196          // (224/16)^2
#define NSP  576          // (384/16)^2
#define BN   (Bb*NSs)     // 32
#define HPp  24           // HS/P
#define CENTER 105        // 7*14+7

typedef __attribute__((ext_vector_type(2))) float v2f;
typedef __attribute__((ext_vector_type(8))) float v8f;

// ---------------- block reductions ----------------
__device__ __forceinline__ float blkMax256(float v, float* red) {
    int t = threadIdx.x;
    red[t] = v; __syncthreads();
    for (int o = 128; o > 0; o >>= 1) { if (t < o) red[t] = fmaxf(red[t], red[t+o]); __syncthreads(); }
    float r = red[0]; __syncthreads(); return r;
}
__device__ __forceinline__ float blkSum256(float v, float* red) {
    int t = threadIdx.x;
    red[t] = v; __syncthreads();
    for (int o = 128; o > 0; o >>= 1) { if (t < o) red[t] += red[t+o]; __syncthreads(); }
    float r = red[0]; __syncthreads(); return r;
}
__device__ __forceinline__ int blkSumI512(int v, int* red) {
    int t = threadIdx.x;
    red[t] = v; __syncthreads();
    for (int o = 256; o > 0; o >>= 1) { if (t < o) red[t] += red[t+o]; __syncthreads(); }
    int r = red[0]; __syncthreads(); return r;
}

// ---------------- 1) normalize template patch features ----------------
// nf[b,n,:] = feats / ||feats||,  feats = template_patches[b, 1+n, :]
__global__ void k_normalize(const float* __restrict__ tp, float* __restrict__ nf) {
    __shared__ float red[256];
    int row = blockIdx.x;                // b*NT + n
    int b = row / NTt, n = row % NTt;
    const float* src = tp + ((size_t)b*(NTt+1) + (n+1))*EMB;
    float s = 0.f;
    for (int d = threadIdx.x; d < EMB; d += 256) { float v = src[d]; s += v*v; }
    float ss = blkSum256(s, red);
    float inv = 1.0f / sqrtf(ss);
    float* dst = nf + ((size_t)b*NTt + n)*EMB;
    for (int d = threadIdx.x; d < EMB; d += 256) dst[d] = src[d]*inv;
}

// ---------------- 2) obj_proto[b,k,d] = sum_n cls_attn[b,k,n]*q_patches[b,n,d] ----------------
// WMMA f32 16x16x4. One wave per (b, 16-wide d-tile). M=12 heads (padded to 16), K=n (196=49*4).
__global__ void k_objproto(const float* __restrict__ attn, const float* __restrict__ query,
                           float* __restrict__ obj) {
    int blk = blockIdx.x;
    int b = blk / (EMB/16), nt = blk % (EMB/16);
    int lane = threadIdx.x;
    int half = lane >> 4, lid = lane & 15;
    int m = lid;                       // A row = head index
    bool mval = m < NHh;
    int d0 = nt*16 + lid;              // B col = embedding index
    int h = d0 >> 6, e = d0 & 63;
    size_t abase = ((size_t)(b*NHh + (mval ? m : 0))) * 197 * 197;   // row 0 of attn
    size_t qbase = ((size_t)(b*NHh + h) * 197) * 64 + e;
    v8f acc = {};
    for (int kk = 0; kk < NTt; kk += 4) {
        int ka = kk + half*2;
        v2f a, bf;
        float a0 = attn[abase + 1 + ka];
        float a1 = attn[abase + 2 + ka];
        a.x = mval ? a0 : 0.f;
        a.y = mval ? a1 : 0.f;
        bf.x = query[qbase + (size_t)(1 + ka)*64];
        bf.y = query[qbase + (size_t)(2 + ka)*64];
        acc = __builtin_amdgcn_wmma_f32_16x16x4_f32(false, a, false, bf,
                                                    (short)0, acc, false, false);
    }
    for (int r = 0; r < 8; ++r) {
        int mm = r + half*8;
        if (mm < NHh) obj[((size_t)b*NHh + mm)*EMB + nt*16 + lid] = acc[r];
    }
}

// ---------------- 3) assignment[b,n,k] = dot(nf[b,n,:], obj_proto[b,k,:]) ----------------
// WMMA f32 16x16x4. One wave per (b, 16-row n-tile). 13 tiles cover 196 rows; N=12 padded to 16.
__global__ void k_assign(const float* __restrict__ nf, const float* __restrict__ obj,
                         float* __restrict__ asg) {
    int blk = blockIdx.x;
    int b = blk / 13, mt = blk % 13;
    int lane = threadIdx.x;
    int half = lane >> 4, lid = lane & 15;
    int m = mt*16 + lid;  bool mval = m < NTt;
    int kc = lid;         bool kval = kc < Kk;
    const float* nfr = nf  + ((size_t)b*NTt + (mval ? m  : 0))*EMB;
    const float* obr = obj + ((size_t)b*NHh + (kval ? kc : 0))*EMB;
    v8f acc = {};
    for (int kk = 0; kk < EMB; kk += 4) {
        int ka = kk + half*2;
        v2f a, bf;
        a.x  = mval ? nfr[ka]   : 0.f;
        a.y  = mval ? nfr[ka+1] : 0.f;
        bf.x = kval ? obr[ka]   : 0.f;
        bf.y = kval ? obr[ka+1] : 0.f;
        acc = __builtin_amdgcn_wmma_f32_16x16x4_f32(false, a, false, bf,
                                                    (short)0, acc, false, false);
    }
    for (int r = 0; r < 8; ++r) {
        int mm = mt*16 + r + half*8;
        if (mm < NTt && lid < Kk) asg[((size_t)b*NTt + mm)*Kk + lid] = acc[r];
    }
}

// ---------------- 4) sinkhorn + argmax + refined prototype ----------------
__global__ void k_sinkhorn(const float* __restrict__ asg, const float* __restrict__ nf,
                           float* __restrict__ refined) {
    __shared__ float sQ[NTt*Kk];
    __shared__ float red[256];
    __shared__ float scol[Kk];
    __shared__ float srow[NTt];
    __shared__ int   stgt;
    int b = blockIdx.x, t = threadIdx.x;
    const float* src = asg + (size_t)b*NTt*Kk;
    for (int i = t; i < NTt*Kk; i += 256) sQ[i] = src[i];
    __syncthreads();
    // global max
    float mx = -1e30f;
    for (int i = t; i < NTt*Kk; i += 256) mx = fmaxf(mx, sQ[i]);
    float M = blkMax256(mx, red);
    // exp((x-m)/eps) and total sum
    float s = 0.f;
    for (int i = t; i < NTt*Kk; i += 256) { float v = expf((sQ[i]-M)*(1.f/0.05f)); sQ[i] = v; s += v; }
    float S = blkSum256(s, red);
    float invS = 1.f / S;
    for (int i = t; i < NTt*Kk; i += 256) sQ[i] *= invS;
    __syncthreads();
    for (int it = 0; it < 3; ++it) {
        if (t < Kk) { float cs = 0.f; for (int n = 0; n < NTt; ++n) cs += sQ[n*Kk + t]; scol[t] = cs; }
        __syncthreads();
        for (int i = t; i < NTt*Kk; i += 256) { int k = i % Kk; sQ[i] /= (scol[k]*Kk); }
        __syncthreads();
        if (t < NTt) { float rs = 0.f; for (int k = 0; k < Kk; ++k) rs += sQ[t*Kk + k]; srow[t] = rs; }
        __syncthreads();
        for (int i = t; i < NTt*Kk; i += 256) { int n = i / Kk; sQ[i] /= (srow[n]*NTt); }
        __syncthreads();
    }
    for (int i = t; i < NTt*Kk; i += 256) sQ[i] *= (float)NTt;
    __syncthreads();
    if (t == 0) {
        int best = 0; float bv = sQ[CENTER*Kk + 0];
        for (int k = 1; k < Kk; ++k) { float v = sQ[CENTER*Kk + k]; if (v > bv) { bv = v; best = k; } }
        stgt = best;
    }
    __syncthreads();
    int tg = stgt;
    for (int d = t; d < EMB; d += 256) {
        float acc = 0.f;
        const float* nfb = nf + (size_t)b*NTt*EMB + d;
        for (int n = 0; n < NTt; ++n) acc += sQ[n*Kk + tg] * nfb[(size_t)n*EMB];
        refined[(size_t)b*EMB + d] = acc;
    }
}

// ---------------- 5) attention matvec + head-mean + softmax -> track_mask ----------------
__global__ void k_attn(const float* __restrict__ refined, const float* __restrict__ skey,
                       float* __restrict__ tmask) {
    __shared__ float sref[EMB];
    __shared__ float satt[NSP];
    __shared__ float red[256];
    int bn = blockIdx.x, t = threadIdx.x;
    int b = bn / NSs;
    for (int i = t; i < EMB; i += 256) sref[i] = refined[(size_t)b*EMB + i];
    __syncthreads();
    for (int n = t; n < NSP; n += 256) {
        float acc = 0.f;
        for (int h = 0; h < NHh; ++h) {
            const float* kp = skey + (((size_t)bn*NHh + h)*577 + 1 + n)*64;
            const float* rp = sref + h*64;
            float pa = 0.f;
            #pragma unroll 8
            for (int e = 0; e < 64; ++e) pa += rp[e]*kp[e];
            acc += pa;
        }
        satt[n] = acc * (1.f/NHh) * 0.125f;   // mean over heads, * HD^-0.5
    }
    __syncthreads();
    float lm = -1e30f;
    for (int n = t; n < NSP; n += 256) lm = fmaxf(lm, satt[n]);
    float M = blkMax256(lm, red);
    float ls = 0.f;
    for (int n = t; n < NSP; n += 256) { float e = expf(satt[n]-M); satt[n] = e; ls += e; }
    float S = blkSum256(ls, red);
    float invS = 1.f / S;
    for (int n = t; n < NSP; n += 256) tmask[(size_t)bn*NSP + n] = satt[n]*invS;
}

// ---------------- 6) on-the-fly 24->384 bilinear upsample, quantile bisection, bbox ----------------
__global__ void k_bbox(const float* __restrict__ tmask, int* __restrict__ bbox) {
    __shared__ float sm[NSP];
    __shared__ int   ai0[HSs];
    __shared__ float af[HSs];
    __shared__ int   ired[512];
    __shared__ int   s_ymin, s_ymax, s_xmin, s_xmax, s_cnt;
    int bn = blockIdx.x, t = threadIdx.x;
    for (int i = t; i < NSP; i += 512) sm[i] = tmask[(size_t)bn*NSP + i];
    for (int i = t; i < HSs; i += 512) {
        float s = (i + 0.5f)*(24.f/384.f) - 0.5f;
        s = fminf(fmaxf(s, 0.f), 23.f);
        int i0 = (int)floorf(s);
        ai0[i] = i0; af[i] = s - (float)i0;
    }
    if (t == 0) { s_ymin = HSs; s_ymax = -1; s_xmin = HSs; s_xmax = -1; s_cnt = 0; }
    __syncthreads();
    // bisection for 0.9 quantile threshold; mask values are in (0,1)
    const int TARGET = 132710;     // ~0.9*(N-1) rounded up, N = 384*384
    float lo = 0.f, hi = 1.f;
    for (int it = 0; it < 22; ++it) {
        float mid = 0.5f*(lo + hi);
        int cnt = 0;
        for (int idx = t; idx < HSs*HSs; idx += 512) {
            int y = idx / HSs, x = idx - y*HSs;
            int y0 = ai0[y]; float fy = af[y]; int y1 = min(y0+1, 23);
            int x0 = ai0[x]; float fx = af[x]; int x1 = min(x0+1, 23);
            float v = (sm[y0*HPp + x0]*(1.f-fy) + sm[y1*HPp + x0]*fy)*(1.f-fx)
                    + (sm[y0*HPp + x1]*(1.f-fy) + sm[y1*HPp + x1]*fy)*fx;
            cnt += (v <= mid) ? 1 : 0;
        }
        int total = blkSumI512(cnt, ired);
        if (total >= TARGET) hi = mid; else lo = mid;
    }
    float thr = hi;
    int lymin = HSs, lymax = -1, lxmin = HSs, lxmax = -1, lcnt = 0;
    for (int idx = t; idx < HSs*HSs; idx += 512) {
        int y = idx / HSs, x = idx - y*HSs;
        int y0 = ai0[y]; float fy = af[y]; int y1 = min(y0+1, 23);
        int x0 = ai0[x]; float fx = af[x]; int x1 = min(x0+1, 23);
        float v = (sm[y0*HPp + x0]*(1.f-fy) + sm[y1*HPp + x0]*fy)*(1.f-fx)
                + (sm[y0*HPp + x1]*(1.f-fy) + sm[y1*HPp + x1]*fy)*fx;
        if (v > thr) {
            lymin = min(lymin, y); lymax = max(lymax, y);
            lxmin = min(lxmin, x); lxmax = max(lxmax, x);
            lcnt++;
        }
    }
    atomicMin(&s_ymin, lymin); atomicMax(&s_ymax, lymax);
    atomicMin(&s_xmin, lxmin); atomicMax(&s_xmax, lxmax);
    atomicAdd(&s_cnt, lcnt);
    __syncthreads();
    if (t == 0) {
        int* o = bbox + bn*5;
        int valid = (s_cnt > 0) ? 1 : 0;
        o[0] = valid ? s_ymin : 0;
        o[1] = valid ? s_xmin : 0;
        o[2] = valid ? s_ymax : 0;
        o[3] = valid ? s_xmax : 0;
        o[4] = valid;
    }
}

// ---------------- 7) crop-resize output ----------------
__device__ __forceinline__ float crop_sample(const float* __restrict__ img, int lim,
                                             int top, int left, int h, int w,
                                             int ty, int tx) {
    // y axis
    float szy = (float)h;
    float sy = (ty + 0.5f)*(szy/(float)TZz) - 0.5f;
    sy = fminf(fmaxf(sy, 0.f), szy - 1.f);
    float fiy = floorf(sy); int iy0 = (int)fiy; float fy = sy - fiy;
    int y0 = min(max(top + iy0, 0), lim - 1);
    int y1 = min(max(min(top + iy0 + 1, top + h - 1), 0), lim - 1);
    // x axis
    float szx = (float)w;
    float sx = (tx + 0.5f)*(szx/(float)TZz) - 0.5f;
    sx = fminf(fmaxf(sx, 0.f), szx - 1.f);
    float fix = floorf(sx); int ix0 = (int)fix; float fx = sx - fix;
    int x0 = min(max(left + ix0, 0), lim - 1);
    int x1 = min(max(min(left + ix0 + 1, left + w - 1), 0), lim - 1);
    float v00 = img[(size_t)y0*lim + x0];
    float v10 = img[(size_t)y1*lim + x0];
    float v01 = img[(size_t)y0*lim + x1];
    float v11 = img[(size_t)y1*lim + x1];
    float r0 = v00*(1.f-fy) + v10*fy;
    float r1 = v01*(1.f-fy) + v11*fy;
    return r0*(1.f-fx) + r1*fx;
}

__global__ void k_crop(const float* __restrict__ frames, const float* __restrict__ tmpl,
                       const int* __restrict__ bbox, float* __restrict__ out) {
    int gid = blockIdx.x*256 + threadIdx.x;
    // gid decodes (bn, c, ty, tx); total = BN*C*TZ*TZ
    int tx = gid & (TZz-1);
    int ty = (gid >> 7) & (TZz-1);
    int c  = (gid >> 14) % Cc;
    int bn = gid / (Cc*TZz*TZz);
    const int* bb = bbox + bn*5;
    float res;
    if (bb[4]) {
        int top = bb[0], left = bb[1];
        int h = max(bb[2]-bb[0], 1);
        int w = max(bb[3]-bb[1], 1);
        const float* img = frames + ((size_t)bn*Cc + c)*HSs*HSs;
        res = crop_sample(img, HSs, top, left, h, w, ty, tx);
    } else {
        int b = bn / NSs;
        const float* img = tmpl + ((size_t)b*Cc + c)*HTt*HTt;
        res = crop_sample(img, HTt, 0, 0, HTt, HTt, ty, tx);
    }
    out[gid] = res;
}

// ---------------- launcher ----------------
extern "C" void kernel_launch(void* const* d_in, const int* in_sizes, int n_in,
                              void* d_out, int out_size, void* d_ws, size_t ws_size,
                              hipStream_t stream) {
    (void)in_sizes; (void)n_in; (void)out_size; (void)ws_size;
    const float* tmpl   = (const float*)d_in[0]; // (B,C,224,224)
    const float* frames = (const float*)d_in[1]; // (B,NS,C,384,384)
    const float* tp     = (const float*)d_in[2]; // (B,197,768)
    const float* tattn  = (const float*)d_in[3]; // (B,12,197,197)
    const float* tquery = (const float*)d_in[4]; // (B,12,197,64)
    const float* skey   = (const float*)d_in[5]; // (32,12,577,64)
    float* out = (float*)d_out;

    char* ws = (char*)d_ws;
    size_t off = 0;
    float* nf      = (float*)(ws + off); off += (size_t)Bb*NTt*EMB*sizeof(float);   // 2.41 MB
    float* obj     = (float*)(ws + off); off += (size_t)Bb*NHh*EMB*sizeof(float);
    float* asg     = (float*)(ws + off); off += (size_t)Bb*NTt*Kk*sizeof(float);
    float* refined = (float*)(ws + off); off += (size_t)Bb*EMB*sizeof(float);
    float* tmask   = (float*)(ws + off); off += (size_t)BN*NSP*sizeof(float);
    int*   bbox    = (int*)  (ws + off); off += (size_t)BN*5*sizeof(int);

    k_normalize<<<Bb*NTt, 256, 0, stream>>>(tp, nf);
    k_objproto <<<Bb*(EMB/16), 32, 0, stream>>>(tattn, tquery, obj);
    k_assign   <<<Bb*13, 32, 0, stream>>>(nf, obj, asg);
    k_sinkhorn <<<Bb, 256, 0, stream>>>(asg, nf, refined);
    k_attn     <<<BN, 256, 0, stream>>>(refined, skey, tmask);
    k_bbox     <<<BN, 512, 0, stream>>>(tmask, bbox);
    k_crop     <<<(BN*Cc*TZz*TZz)/256, 256, 0, stream>>>(frames, tmpl, bbox, out);
}